// TestModel_13477607375385
// MI455X (gfx1250) — compile-verified
//
#include <hip/hip_runtime.h>

// ---------------------------------------------------------------------------
// Problem constants (match reference: T,N,D,B,L = 4, 1e6, 4, 16384, 50)
// ---------------------------------------------------------------------------
constexpr int T      = 4;
constexpr int NROWS  = 1000000;
constexpr int D      = 4;
constexpr int BAGS   = 16384;
constexpr int TOTAL  = 16384 * 50;   // 819200 lookups per table
constexpr int CHUNK  = 64;           // consecutive lookups per thread

typedef float v2f __attribute__((ext_vector_type(2)));
typedef float v8f __attribute__((ext_vector_type(8)));
typedef int   v4i __attribute__((ext_vector_type(4)));

// ---------------------------------------------------------------------------
// Kernel 1: zero the pooled scratch [T*BAGS, 4] floats (harness poisons d_ws)
// ---------------------------------------------------------------------------
__global__ void k_zero_ws(float4* __restrict__ ws, int n4) {
    int i = blockIdx.x * blockDim.x + threadIdx.x;
    if (i < n4) ws[i] = make_float4(0.f, 0.f, 0.f, 0.f);
}

// ---------------------------------------------------------------------------
// Kernel 2: per-table gather + segment-sum pooling.
// Each thread owns CHUNK consecutive (sorted-segment) lookups; accumulates a
// float4 in registers and flushes with global_atomic_add_f32 only on segment
// boundaries (avg ~1.3 flushes per thread). int4 loads for indices/segids,
// float4 (b128) loads for the 16-byte embedding rows (L2-resident tables).
// ---------------------------------------------------------------------------
__device__ __forceinline__ void flush_bag(float* __restrict__ pout, int seg,
                                          const float4& a) {
    atomicAdd(pout + seg * 4 + 0, a.x);
    atomicAdd(pout + seg * 4 + 1, a.y);
    atomicAdd(pout + seg * 4 + 2, a.z);
    atomicAdd(pout + seg * 4 + 3, a.w);
}

__global__ void k_pool(const float* __restrict__ tables,
                       const int*   __restrict__ indices,
                       const int*   __restrict__ segids,
                       float*       __restrict__ pooled /* [T*BAGS*4] */) {
    constexpr int CPT = TOTAL / CHUNK;            // chunks per table (12800)
    int tid = blockIdx.x * blockDim.x + threadIdx.x;
    if (tid >= T * CPT) return;
    const int t    = tid / CPT;
    const int c    = tid % CPT;
    const int base = c * CHUNK;

    const v4i* __restrict__ idx4 =
        (const v4i*)(indices + (size_t)t * TOTAL + base);
    const v4i* __restrict__ seg4 =
        (const v4i*)(segids  + (size_t)t * TOTAL + base);
    const float4* __restrict__ tbl =
        (const float4*)(tables + (size_t)t * NROWS * D);
    float* __restrict__ pout = pooled + (size_t)t * BAGS * D;

    float4 acc = make_float4(0.f, 0.f, 0.f, 0.f);
    int cur = seg4[0][0];

    for (int q = 0; q < CHUNK / 4; ++q) {
        v4i s4 = seg4[q];
        v4i i4 = idx4[q];
#pragma unroll
        for (int k = 0; k < 4; ++k) {
            int s = s4[k];
            if (s != cur) {                        // sorted => rare divergence
                flush_bag(pout, cur, acc);
                acc = make_float4(0.f, 0.f, 0.f, 0.f);
                cur = s;
            }
            float4 r = tbl[i4[k]];                 // global_load_b128 gather
            acc.x += r.x; acc.y += r.y; acc.z += r.z; acc.w += r.w;
        }
    }
    flush_bag(pout, cur, acc);
}

// ---------------------------------------------------------------------------
// Kernel 3: over-arch Linear(4,1) via V_WMMA_F32_16X16X4_F32 (exact f32 path).
//   A (16x4): every row = W           -> lane<16: (W0,W1), lane>=16: (W2,W3)
//   B (4x16): column n = pooled row n -> lane<16: row[lane].k01,
//                                        lane>=16: row[lane-16].k23
//   D[0][n] = dot(W, pooled_row n): C/D VGPR0 lane n (n<16) holds out row n.
// One wave -> 16 outputs; EXEC all-ones through the WMMA.
// ---------------------------------------------------------------------------
__global__ void k_linear_wmma(const float* __restrict__ pooled,
                              const float* __restrict__ W,
                              const float* __restrict__ bias_p,
                              float*       __restrict__ out) {
    const int lane    = threadIdx.x & 31;
    const int wave    = threadIdx.x >> 5;
    const int rowBase = (blockIdx.x * (blockDim.x >> 5) + wave) * 16;
    const bool hi     = lane >= 16;

    // A matrix: broadcast weight vector into every row.
    v2f a;
    a.x = hi ? W[2] : W[0];
    a.y = hi ? W[3] : W[1];

    // B matrix: 16 pooled rows as columns (K-halves split across lane halves).
    const int r = rowBase + (lane & 15);
    const float* prow = pooled + (size_t)r * 4 + (hi ? 2 : 0);
    v2f bm;
    bm.x = prow[0];
    bm.y = prow[1];

    v8f c = {};
    c = __builtin_amdgcn_wmma_f32_16x16x4_f32(
            /*neg_a=*/false, a, /*neg_b=*/false, bm,
            /*c_mod=*/(short)0, c, /*reuse_a=*/false, /*reuse_b=*/false);

    if (lane < 16) out[rowBase + lane] = c[0] + bias_p[0];
}

// ---------------------------------------------------------------------------
// Launch: zero ws -> pool (atomics into ws) -> wmma linear (ws -> out)
// ---------------------------------------------------------------------------
extern "C" void kernel_launch(void* const* d_in, const int* in_sizes, int n_in,
                              void* d_out, int out_size, void* d_ws,
                              size_t ws_size, hipStream_t stream) {
    const float* tables  = (const float*)d_in[0];  // [T, N, D] f32
    const float* W       = (const float*)d_in[1];  // [1, D]    f32
    const float* bias    = (const float*)d_in[2];  // [1]       f32
    const int*   indices = (const int*)d_in[3];    // [T, TOTAL] i32
    const int*   segids  = (const int*)d_in[4];    // [T, TOTAL] i32 (sorted)
    float*       out     = (float*)d_out;          // [T*BAGS, 1] f32
    float*       pooled  = (float*)d_ws;           // scratch [T*BAGS*4] f32

    // 1) zero pooled scratch (T*BAGS float4 = 65536)
    {
        int n4 = T * BAGS;
        k_zero_ws<<<(n4 + 255) / 256, 256, 0, stream>>>((float4*)pooled, n4);
    }
    // 2) gather + segmented sum: T * (TOTAL/CHUNK) = 51200 threads
    {
        int nthreads = T * (TOTAL / CHUNK);
        k_pool<<<(nthreads + 255) / 256, 256, 0, stream>>>(tables, indices,
                                                           segids, pooled);
    }
    // 3) WMMA linear: T*BAGS/16 = 4096 waves, 8 waves/block -> 512 blocks
    {
        int waves  = (T * BAGS) / 16;
        int blocks = waves / 8;
        k_linear_wmma<<<blocks, 256, 0, stream>>>(pooled, W, bias, out);
    }
}